// DecoderRNN_50543175139653
// MI455X (gfx1250) — compile-verified
//
#include <hip/hip_runtime.h>
#include <hip/hip_bf16.h>
#include <math.h>

typedef __attribute__((ext_vector_type(2))) float v2f;
typedef __attribute__((ext_vector_type(4))) float v4f;
typedef __attribute__((ext_vector_type(8))) float v8f;

#define HID 1024
#define G4H 4096
#define VOC 32000
#define BAT 64
#define TLEN 60

// Projection-kernel tiling
#define KC 64        // K chunk staged in LDS per iteration
#define LDST 68      // padded LDS row stride in floats (272B: 16B aligned, bank-conflict-free)
#define PROJ_ASYNC 1 // 1: gfx1250 global_load_async_to_lds_b128 DMA; 0: ds_store fallback

__device__ __forceinline__ float sigmf(float x) { return 1.0f / (1.0f + expf(-x)); }

__device__ __forceinline__ v8f wmma_f32(v2f a, v2f b, v8f c) {
  return __builtin_amdgcn_wmma_f32_16x16x4_f32(false, a, false, b, (short)0, c,
                                               false, false);
}

// Stage 16 bytes global -> LDS (async DMA on gfx1250, sync fallback otherwise).
__device__ __forceinline__ void stage16(const float* __restrict__ gp, float* lp) {
#if PROJ_ASYNC
  asm volatile("global_load_async_to_lds_b128 %0, %1, off"
               :
               : "v"((unsigned)(unsigned long long)lp),
                 "v"((unsigned long long)gp)
               : "memory");
#else
  *(v4f*)lp = *(const v4f*)gp;
#endif
}

__device__ __forceinline__ void stage_wait_and_barrier() {
#if PROJ_ASYNC
  asm volatile("s_wait_asynccnt 0x0" ::: "memory");
#endif
  __syncthreads();
}

// ---------------------------------------------------------------------------
// Kernel 1: Gx[t*B+b, :] = relu(emb[token(t,b)]) @ W_ih^T + (b_ih + b_hh)
// M = T*B = 3840, N = 4096, K = 1024.  1 wave = 1 M-tile x 4 N-tiles (16x64).
// grid: 3840 blocks x 128 threads (4 waves) -> 15360 waves = 240 * 64.
// ---------------------------------------------------------------------------
__global__ void k_embed_gates(const float* __restrict__ emb,
                              const int* __restrict__ target,
                              const float* __restrict__ W_ih,
                              const float* __restrict__ b_ih,
                              const float* __restrict__ b_hh,
                              float* __restrict__ Gx) {
  const int lane   = threadIdx.x & 31;
  const int waveId = blockIdx.x * (blockDim.x >> 5) + (threadIdx.x >> 5);
  const int mt = waveId >> 6;   // 0..239  (16-row M tile)
  const int ng = waveId & 63;   // 0..63   (64-col N group)

  const int arow = mt * 16 + (lane & 15);
  const int t    = arow >> 6;          // row = t*64 + b
  const int b    = arow & 63;
  const int tok  = (t == 0) ? 0 : target[b * TLEN + (t - 1)];
  const float* __restrict__ ap = emb + (size_t)tok * HID;
  const int kl = (lane >> 4) << 1;     // K sub-offset 0 or 2

  const int ncol0 = ng * 64 + (lane & 15);

  v8f acc[4] = {v8f{}, v8f{}, v8f{}, v8f{}};

  for (int k = 0; k < HID; k += 4) {
    const int k0 = k + kl;
    v2f a;
    a.x = fmaxf(ap[k0], 0.0f);
    a.y = fmaxf(ap[k0 + 1], 0.0f);
    const float* __restrict__ wp = W_ih + (size_t)ncol0 * HID + k0;
#pragma unroll
    for (int nt = 0; nt < 4; ++nt) {
      v2f bm;
      bm.x = wp[0];
      bm.y = wp[1];
      acc[nt] = wmma_f32(a, bm, acc[nt]);
      wp += (size_t)16 * HID;
    }
  }

#pragma unroll
  for (int nt = 0; nt < 4; ++nt) {
    const int col  = ng * 64 + nt * 16 + (lane & 15);
    const float bias = b_ih[col] + b_hh[col];
#pragma unroll
    for (int r = 0; r < 8; ++r) {
      const int row = mt * 16 + r + ((lane >> 4) << 3);
      Gx[(size_t)row * G4H + col] = acc[nt][r] + bias;
    }
  }
}

// ---------------------------------------------------------------------------
// Kernel 2: one LSTM time step (fused recurrent GEMM + gate nonlinearities).
// Wave owns (b-tile, h-col tile), computes all 4 gate tiles for those columns
// so the c/h update is local.  h is double-buffered across launches.
// grid: 64 blocks x 128 threads -> 256 waves.
// ---------------------------------------------------------------------------
__global__ void k_lstm_step(const float* __restrict__ W_hh,
                            const float* __restrict__ Gx, int t,
                            const float* __restrict__ h_in,
                            float* __restrict__ h_out,
                            float* __restrict__ cbuf,
                            float* __restrict__ hs) {
  const int lane   = threadIdx.x & 31;
  const int waveId = blockIdx.x * (blockDim.x >> 5) + (threadIdx.x >> 5);
  const int mt = waveId >> 6;   // 0..3   (16-row b tile)
  const int nt = waveId & 63;   // 0..63  (16-col h tile)

  const int arow = mt * 16 + (lane & 15);
  const float* __restrict__ ap = h_in + (size_t)arow * HID;
  const int kl = (lane >> 4) << 1;
  const int ncol = nt * 16 + (lane & 15);

  v8f acc[4] = {v8f{}, v8f{}, v8f{}, v8f{}};   // i, f, g, o tiles

  for (int k = 0; k < HID; k += 4) {
    const int k0 = k + kl;
    v2f a;
    a.x = ap[k0];
    a.y = ap[k0 + 1];
#pragma unroll
    for (int g = 0; g < 4; ++g) {
      const float* __restrict__ wp = W_hh + (size_t)(g * HID + ncol) * HID + k0;
      v2f bm;
      bm.x = wp[0];
      bm.y = wp[1];
      acc[g] = wmma_f32(a, bm, acc[g]);
    }
  }

  const float* __restrict__ gx = Gx + (size_t)(t * BAT) * G4H;
#pragma unroll
  for (int r = 0; r < 8; ++r) {
    const int brow = mt * 16 + r + ((lane >> 4) << 3);
    const int col  = nt * 16 + (lane & 15);
    const size_t gbase = (size_t)brow * G4H + col;
    const float gi = sigmf(acc[0][r] + gx[gbase]);
    const float gf = sigmf(acc[1][r] + gx[gbase + HID]);
    const float gg = tanhf(acc[2][r] + gx[gbase + 2 * HID]);
    const float go = sigmf(acc[3][r] + gx[gbase + 3 * HID]);
    const size_t hidx = (size_t)brow * HID + col;
    const float cnew = gf * cbuf[hidx] + gi * gg;
    const float hnew = go * tanhf(cnew);
    cbuf[hidx]  = cnew;
    h_out[hidx] = hnew;
    hs[(size_t)(t * BAT + brow) * HID + col] = hnew;
  }
}

// ---------------------------------------------------------------------------
// Kernel 3: logits = hs @ W_out^T + b_out -> d_out in [B,T,V] order.
// M = 3840, N = 32000, K = 1024.
// Block = 8 waves = 128 M-rows x 64 N-cols; B tiles (W_out) staged in LDS and
// shared by all 8 waves (cuts W_out L2 traffic 8x vs per-wave streaming).
// Staging is double-buffered async DMA (global_load_async_to_lds_b128) that
// overlaps the WMMA loop of the previous K-chunk.
// grid: 30 * 500 = 15000 blocks x 256 threads.
// ---------------------------------------------------------------------------
__global__ void k_projection(const float* __restrict__ hs,
                             const float* __restrict__ W_out,
                             const float* __restrict__ b_out,
                             float* __restrict__ out) {
  __shared__ float bsh[2][64 * LDST];   // 2 x 17.4 KB

  const int lane = threadIdx.x & 31;
  const int wave = threadIdx.x >> 5;    // 0..7 -> M sub-tile
  const int mb = blockIdx.x / 500;      // 0..29  (128-row M block)
  const int nb = blockIdx.x % 500;      // 0..499 (64-col N group)

  const int arow = mb * 128 + wave * 16 + (lane & 15);
  const float* __restrict__ pa = hs + (size_t)arow * HID;
  const int kl = (lane >> 4) << 1;

  // Staging slots: thread -> (base col, float4 offset) inside the 64xKC chunk.
  const int sc = threadIdx.x >> 4;         // 0..15 base column
  const int ss = (threadIdx.x & 15) << 2;  // float offset 0..60 step 4
  const float* __restrict__ gW = W_out + (size_t)(nb * 64 + sc) * HID + ss;

  v8f acc[4] = {v8f{}, v8f{}, v8f{}, v8f{}};

  // Prologue: stage chunk 0 into buffer 0.
#pragma unroll
  for (int cc = 0; cc < 64; cc += 16)
    stage16(gW + (size_t)cc * HID, &bsh[0][(sc + cc) * LDST + ss]);

  const int NCHUNK = HID / KC;   // 16
  for (int ci = 0; ci < NCHUNK; ++ci) {
    stage_wait_and_barrier();    // chunk ci resident; prev buffer free

    if (ci + 1 < NCHUNK) {
      const int knext = (ci + 1) * KC;
      float* dst = bsh[(ci + 1) & 1];
#pragma unroll
      for (int cc = 0; cc < 64; cc += 16)
        stage16(gW + (size_t)cc * HID + knext, &dst[(sc + cc) * LDST + ss]);
    }

    const float* __restrict__ bbuf = bsh[ci & 1];
    const int kc = ci * KC;
#pragma unroll 4
    for (int k = 0; k < KC; k += 4) {
      const int k0 = k + kl;
      v2f a;
      a.x = pa[kc + k0];
      a.y = pa[kc + k0 + 1];
#pragma unroll
      for (int nt = 0; nt < 4; ++nt) {
        const v2f bm = *(const v2f*)&bbuf[(nt * 16 + (lane & 15)) * LDST + k0];
        acc[nt] = wmma_f32(a, bm, acc[nt]);
      }
    }
  }

#pragma unroll
  for (int nt = 0; nt < 4; ++nt) {
    const int col  = nb * 64 + nt * 16 + (lane & 15);
    const float bias = b_out[col];
#pragma unroll
    for (int r = 0; r < 8; ++r) {
      const int m  = mb * 128 + wave * 16 + r + ((lane >> 4) << 3);
      const int tt = m >> 6;          // hs rows are (t, b)
      const int bb = m & 63;
      out[(size_t)(bb * TLEN + tt) * VOC + col] = acc[nt][r] + bias;
    }
  }
}

// ---------------------------------------------------------------------------
// Kernel 4: in-place log_softmax over V for each of the B*T rows of d_out.
// One 256-thread workgroup per row; LDS tree reductions (wave32-safe).
// ---------------------------------------------------------------------------
__global__ void k_logsoftmax(float* __restrict__ logits) {
  __shared__ float red[256];
  float* __restrict__ p = logits + (size_t)blockIdx.x * VOC;
  const int tid = threadIdx.x;

  float m = -INFINITY;
  for (int i = tid; i < VOC; i += 256) m = fmaxf(m, p[i]);
  red[tid] = m;
  __syncthreads();
  for (int s = 128; s > 0; s >>= 1) {
    if (tid < s) red[tid] = fmaxf(red[tid], red[tid + s]);
    __syncthreads();
  }
  const float rowmax = red[0];
  __syncthreads();

  float sum = 0.0f;
  for (int i = tid; i < VOC; i += 256) sum += expf(p[i] - rowmax);
  red[tid] = sum;
  __syncthreads();
  for (int s = 128; s > 0; s >>= 1) {
    if (tid < s) red[tid] += red[tid + s];
    __syncthreads();
  }
  const float lse = rowmax + logf(red[0]);
  __syncthreads();

  for (int i = tid; i < VOC; i += 256) p[i] -= lse;
}

// ---------------------------------------------------------------------------
extern "C" void kernel_launch(void* const* d_in, const int* in_sizes, int n_in,
                              void* d_out, int out_size, void* d_ws,
                              size_t ws_size, hipStream_t stream) {
  (void)in_sizes; (void)n_in; (void)out_size; (void)ws_size;
  // Inputs per setup_inputs() order:
  // 0 encoder_outputs (unused), 1 h0, 2 c0, 3 target, 4 emb,
  // 5 W_ih, 6 W_hh, 7 b_ih, 8 b_hh, 9 W_out, 10 b_out
  const float* h0     = (const float*)d_in[1];
  const float* c0     = (const float*)d_in[2];
  const int*   target = (const int*)d_in[3];
  const float* emb    = (const float*)d_in[4];
  const float* W_ih   = (const float*)d_in[5];
  const float* W_hh   = (const float*)d_in[6];
  const float* b_ih   = (const float*)d_in[7];
  const float* b_hh   = (const float*)d_in[8];
  const float* W_out  = (const float*)d_in[9];
  const float* b_out  = (const float*)d_in[10];
  float* out = (float*)d_out;

  // Workspace layout (floats): Gx | hs | h_buf0 | h_buf1 | c_buf
  float* Gx  = (float*)d_ws;
  float* hs  = Gx + (size_t)TLEN * BAT * G4H;       // 15,728,640 floats
  float* hb0 = hs + (size_t)TLEN * BAT * HID;       //  3,932,160 floats
  float* hb1 = hb0 + (size_t)BAT * HID;
  float* cb  = hb1 + (size_t)BAT * HID;

  hipMemcpyAsync(hb0, h0, (size_t)BAT * HID * sizeof(float),
                 hipMemcpyDeviceToDevice, stream);
  hipMemcpyAsync(cb, c0, (size_t)BAT * HID * sizeof(float),
                 hipMemcpyDeviceToDevice, stream);

  // 1) All input-side gate pre-activations in one big WMMA GEMM.
  k_embed_gates<<<3840, 128, 0, stream>>>(emb, target, W_ih, b_ih, b_hh, Gx);

  // 2) Sequential recurrent scan, h double-buffered across launches.
  for (int t = 0; t < TLEN; ++t) {
    const float* h_in = (t & 1) ? hb1 : hb0;
    float* h_out      = (t & 1) ? hb0 : hb1;
    k_lstm_step<<<64, 128, 0, stream>>>(W_hh, Gx, t, h_in, h_out, cb, hs);
  }
  // T=60 even -> final h lives in hb0.

  // 3) Output projection (dominant GEMM), logits straight into d_out.
  k_projection<<<15000, 256, 0, stream>>>(hs, W_out, b_out, out);

  // 4) In-place row-wise log_softmax.
  k_logsoftmax<<<BAT * TLEN, 256, 0, stream>>>(out);

  // 5) Append h_T and c_T after the log-prob block.
  float* out_h = out + (size_t)BAT * TLEN * VOC;
  float* out_c = out_h + (size_t)BAT * HID;
  hipMemcpyAsync(out_h, hb0, (size_t)BAT * HID * sizeof(float),
                 hipMemcpyDeviceToDevice, stream);
  hipMemcpyAsync(out_c, cb, (size_t)BAT * HID * sizeof(float),
                 hipMemcpyDeviceToDevice, stream);
}